// WeightedSAGEConv_420906795210
// MI455X (gfx1250) — compile-verified
//
#include <hip/hip_runtime.h>

typedef __attribute__((ext_vector_type(2))) float v2f;
typedef __attribute__((ext_vector_type(8))) float v8f;

#define NN 50000
#define DD 64
#define NE 800000

// ---------------- zero workspace ----------------
__global__ void ws_zero_kernel(float* __restrict__ p, int n) {
    int i = blockIdx.x * blockDim.x + threadIdx.x;
    if (i < n) p[i] = 0.0f;
}

// ---------------- edge scatter: msg = feat[src]*w -> atomic add to neigh_sum[dst]; deg count ----------------
__global__ __launch_bounds__(256) void edge_scatter_kernel(
    const float* __restrict__ feat, const float* __restrict__ w,
    const int* __restrict__ src, const int* __restrict__ dst,
    float* __restrict__ nsum, float* __restrict__ deg) {
    int tid = blockIdx.x * blockDim.x + threadIdx.x;
    int e = tid >> 4;        // 16 threads per edge, 4 floats each
    int c = tid & 15;
    if (e >= NE) return;
    int s = src[e];
    int d = dst[e];
    float wt = w[e];
    const float4 f = *(const float4*)(feat + (size_t)s * DD + c * 4);
    float* b = nsum + (size_t)d * DD + c * 4;
    atomicAdd(b + 0, f.x * wt);
    atomicAdd(b + 1, f.y * wt);
    atomicAdd(b + 2, f.z * wt);
    atomicAdd(b + 3, f.w * wt);
    if (c == 0) atomicAdd(deg + d, 1.0f);
}

// ---------------- fused dual-GEMM via f32 WMMA ----------------
// out = feat @ W_self^T + b_self + (nsum/max(deg,1)) @ W_neigh^T + b_neigh
//
// LDS holds W^T in pair-major layout so each lane's B operand
// {B[k][n], B[k+1][n]} = {W[n][k], W[n][k+1]} is ONE ds_load_b64 into an
// even-aligned VGPR pair (no repacking movs, bank-conflict-free).
__global__ __launch_bounds__(256) void sage_gemm_wmma_kernel(
    const float* __restrict__ feat,
    const float* __restrict__ Wself, const float* __restrict__ bself,
    const float* __restrict__ Wneigh, const float* __restrict__ bneigh,
    const float* __restrict__ nsum, const float* __restrict__ deg,
    float* __restrict__ out) {
    // ldsW2[p*64 + n] = { W[n][2p], W[n][2p+1] },  p = 0..31, n = 0..63
    __shared__ float2 ldsWs2[(DD / 2) * DD];
    __shared__ float2 ldsWn2[(DD / 2) * DD];
    int tid = threadIdx.x;
    for (int i = tid; i < (DD / 2) * DD; i += 256) {
        int p = i & 31;          // consecutive threads -> consecutive pairs in a row (coalesced)
        int n = i >> 5;
        float2 vs = *(const float2*)(Wself + n * DD + 2 * p);
        float2 vn = *(const float2*)(Wneigh + n * DD + 2 * p);
        ldsWs2[p * DD + n] = vs;
        ldsWn2[p * DD + n] = vn;
    }
    __syncthreads();

    int wave = tid >> 5;     // 8 waves per block, 16 nodes each
    int lane = tid & 31;
    int h = lane >> 4;       // K-half (A) / M-half (D) selector
    int m = lane & 15;       // A row within tile; B/D column within tile
    int node_base = blockIdx.x * 128 + wave * 16;
    if (node_base >= NN) return;   // wave-uniform: EXEC stays all-1s for WMMA

    int arow = node_base + m;
    float invd = 1.0f / fmaxf(deg[arow], 1.0f);
    const float* fr = feat + (size_t)arow * DD;
    const float* nr = nsum + (size_t)arow * DD;

    v8f acc[4] = {v8f{}, v8f{}, v8f{}, v8f{}};

#pragma unroll
    for (int t = 0; t < 16; ++t) {
        int k0 = t * 4 + h * 2;            // this lane's K pair for A VGPRs 0,1
        int p = 2 * t + h;                 // pair index in LDS
        float2 as2 = *(const float2*)(fr + k0);
        float2 an2 = *(const float2*)(nr + k0);
        v2f aS; aS[0] = as2.x;        aS[1] = as2.y;
        v2f aN; aN[0] = an2.x * invd; aN[1] = an2.y * invd;
        const float2* bsrow = ldsWs2 + p * DD;
        const float2* bnrow = ldsWn2 + p * DD;
#pragma unroll
        for (int ct = 0; ct < 4; ++ct) {
            int n = ct * 16 + m;
            v2f bS = *(const v2f*)(bsrow + n);
            v2f bN = *(const v2f*)(bnrow + n);
            acc[ct] = __builtin_amdgcn_wmma_f32_16x16x4_f32(
                false, aS, false, bS, (short)0, acc[ct], false, false);
            acc[ct] = __builtin_amdgcn_wmma_f32_16x16x4_f32(
                false, aN, false, bN, (short)0, acc[ct], false, false);
        }
    }

    // D layout: VGPR j -> row (8*h + j), col = ct*16 + m. Add biases at store.
#pragma unroll
    for (int ct = 0; ct < 4; ++ct) {
        int n = ct * 16 + m;
        float bias = bself[n] + bneigh[n];
#pragma unroll
        for (int j = 0; j < 8; ++j) {
            int row = node_base + h * 8 + j;
            out[(size_t)row * DD + n] = acc[ct][j] + bias;
        }
    }
}

extern "C" void kernel_launch(void* const* d_in, const int* in_sizes, int n_in,
                              void* d_out, int out_size, void* d_ws, size_t ws_size,
                              hipStream_t stream) {
    const float* feat   = (const float*)d_in[0];
    const float* weight = (const float*)d_in[1];
    const float* Wself  = (const float*)d_in[2];
    const float* bself  = (const float*)d_in[3];
    const float* Wneigh = (const float*)d_in[4];
    const float* bneigh = (const float*)d_in[5];
    const int*   src    = (const int*)d_in[6];
    const int*   dst    = (const int*)d_in[7];
    float* out = (float*)d_out;

    float* nsum = (float*)d_ws;                       // NN*DD floats
    float* deg  = nsum + (size_t)NN * DD;             // NN floats

    int ztot = NN * DD + NN;
    ws_zero_kernel<<<(ztot + 255) / 256, 256, 0, stream>>>(nsum, ztot);

    int sthreads = NE * 16;
    edge_scatter_kernel<<<sthreads / 256, 256, 0, stream>>>(feat, weight, src, dst, nsum, deg);

    sage_gemm_wmma_kernel<<<(NN + 127) / 128, 256, 0, stream>>>(
        feat, Wself, bself, Wneigh, bneigh, nsum, deg, out);
}